// DensityGCNProcessor_50989851738542
// MI455X (gfx1250) — compile-verified
//
#include <hip/hip_runtime.h>
#include <hip/hip_bf16.h>
#include <float.h>

typedef __attribute__((ext_vector_type(2))) float v2f;
typedef __attribute__((ext_vector_type(8))) float v8f;

#define NN    4096          // nodes per batch (64x64)
#define BATCH 4
#define NTOT  (NN*BATCH)    // 16384
#define KNN   4
#define ETOT  (NTOT*(KNN+1)) // 81920 edges incl. self loops
#define CIN   256
#define CHID  512
#define COUT  256

// ---------------------------------------------------------------- init
__global__ void init_deg_kernel(int* __restrict__ indeg, int* __restrict__ cursor) {
  int i = blockIdx.x * blockDim.x + threadIdx.x;
  if (i < NTOT) { indeg[i] = 1; cursor[i] = 0; }   // self loop counts toward deg
}

// ---------------------------------------------------------------- kNN on density
// One thread per node; batch density slice staged in LDS (16 KB of 320 KB/WGP).
// Stable tie-break (ascending j + strict '<') matches jnp.argsort semantics.
__global__ void knn_kernel(const float* __restrict__ dens,
                           int* __restrict__ tgt, int* __restrict__ indeg) {
  __shared__ float sd[NN];
  int node = blockIdx.x * blockDim.x + threadIdx.x;   // blocks never cross batches
  int b = node / NN;
  const float* dsrc = dens + b * NN;
  for (int j = threadIdx.x; j < NN; j += blockDim.x) sd[j] = dsrc[j];
  __syncthreads();
  int i = node % NN;
  float di = sd[i];
  float d0 = FLT_MAX, d1 = FLT_MAX, d2 = FLT_MAX, d3 = FLT_MAX;
  int   i0 = 0, i1 = 0, i2 = 0, i3 = 0;
  for (int j = 0; j < NN; ++j) {
    float dist = fabsf(sd[j] - di);
    if (j == i) continue;
    if (dist < d3) {
      if (dist < d0)      { d3=d2;i3=i2; d2=d1;i2=i1; d1=d0;i1=i0; d0=dist;i0=j; }
      else if (dist < d1) { d3=d2;i3=i2; d2=d1;i2=i1; d1=dist;i1=j; }
      else if (dist < d2) { d3=d2;i3=i2; d2=dist;i2=j; }
      else                { d3=dist;i3=j; }
    }
  }
  int base = b * NN;
  int t0 = base + i0, t1 = base + i1, t2 = base + i2, t3 = base + i3;
  tgt[node*KNN+0] = t0; tgt[node*KNN+1] = t1;
  tgt[node*KNN+2] = t2; tgt[node*KNN+3] = t3;
  atomicAdd(&indeg[t0], 1); atomicAdd(&indeg[t1], 1);
  atomicAdd(&indeg[t2], 1); atomicAdd(&indeg[t3], 1);
}

// ---------------------------------------------------------------- prefix scan (16384 -> rowptr)
__global__ void scan_kernel(const int* __restrict__ indeg, int* __restrict__ rowptr) {
  __shared__ int part[1024];
  int t = threadIdx.x;
  int base = t * 16;
  int loc[16];
  int s = 0;
  #pragma unroll
  for (int i = 0; i < 16; ++i) { loc[i] = s; s += indeg[base + i]; }
  part[t] = s;
  __syncthreads();
  for (int off = 1; off < 1024; off <<= 1) {
    int v = 0;
    if (t >= off) v = part[t - off];
    __syncthreads();
    if (t >= off) part[t] += v;
    __syncthreads();
  }
  int prefix = (t == 0) ? 0 : part[t - 1];
  #pragma unroll
  for (int i = 0; i < 16; ++i) rowptr[base + i] = prefix + loc[i];
  if (t == 1023) rowptr[NTOT] = part[1023];
}

// ---------------------------------------------------------------- deg^-1/2
__global__ void dinv_kernel(const int* __restrict__ indeg, float* __restrict__ dinv) {
  int i = blockIdx.x * blockDim.x + threadIdx.x;
  if (i < NTOT) dinv[i] = rsqrtf((float)indeg[i]);
}

// ---------------------------------------------------------------- CSR fill (reverse adjacency, per-edge norm)
__global__ void fill_kernel(const int* __restrict__ tgt, const int* __restrict__ rowptr,
                            int* __restrict__ cursor, const float* __restrict__ dinv,
                            int* __restrict__ eSrc, float* __restrict__ eW) {
  int i = blockIdx.x * blockDim.x + threadIdx.x;
  if (i >= NTOT) return;
  float di = dinv[i];
  int pos = rowptr[i] + atomicAdd(&cursor[i], 1);       // self loop
  eSrc[pos] = i; eW[pos] = di * di;
  #pragma unroll
  for (int k = 0; k < KNN; ++k) {
    int t = tgt[i*KNN + k];
    int p = rowptr[t] + atomicAdd(&cursor[t], 1);
    eSrc[p] = i; eW[p] = di * dinv[t];
  }
}

// ---------------------------------------------------------------- fp32 WMMA GEMM: D[M,Nc] = A[M,K] * W[K,Nc]
// One wave -> 32(M) x 64(N) tile via 8 accumulators; K-loop step 4.
// COLMAJOR: A element (m,k) at A[b*K*NN + k*NN + n] (feature_maps BCHW layout).
// Otherwise A row-major -> K-adjacent pair loads as one b64.
template<bool COLMAJOR>
__global__ void gemm32x64_wmma(const float* __restrict__ A, const float* __restrict__ W,
                               float* __restrict__ D, int Ktot, int Nc) {
  const int lane = threadIdx.x & 31;
  const int wave = threadIdx.x >> 5;
  const int row  = lane & 15;
  const int koff = (lane >> 4) << 1;        // lanes 0-15: K0/K1, lanes 16-31: K2/K3
  const int nodeBase = blockIdx.x << 5;     // 32-row M tile (never crosses batch: 4096%32==0)
  const int colBase  = wave * 64;

  const float* aLo;                         // rows nodeBase .. +15
  const float* aHi;                         // rows nodeBase+16 .. +31
  if (COLMAJOR) {
    int b = nodeBase >> 12;                 // /NN
    int n = (nodeBase & (NN - 1)) + row;
    aLo = A + (size_t)b * Ktot * NN + n;
    aHi = aLo + 16;
  } else {
    aLo = A + (size_t)(nodeBase + row) * Ktot;
    aHi = aLo + (size_t)16 * Ktot;
  }
  const int col = colBase + row;

  v8f accL0 = {}, accL1 = {}, accL2 = {}, accL3 = {};
  v8f accH0 = {}, accH1 = {}, accH2 = {}, accH3 = {};
  for (int k0 = 0; k0 < Ktot; k0 += 4) {
    v2f aL, aH;
    if (COLMAJOR) {
      aL.x = aLo[(size_t)(k0 + koff)     * NN];
      aL.y = aLo[(size_t)(k0 + koff + 1) * NN];
      aH.x = aHi[(size_t)(k0 + koff)     * NN];
      aH.y = aHi[(size_t)(k0 + koff + 1) * NN];
    } else {
      aL = *(const v2f*)(aLo + k0 + koff);  // global_load_b64
      aH = *(const v2f*)(aHi + k0 + koff);
    }
    const float* w0 = W + (size_t)(k0 + koff)     * Nc;
    const float* w1 = W + (size_t)(k0 + koff + 1) * Nc;
    v2f b0; b0.x = w0[col];      b0.y = w1[col];
    v2f b1; b1.x = w0[col + 16]; b1.y = w1[col + 16];
    v2f b2; b2.x = w0[col + 32]; b2.y = w1[col + 32];
    v2f b3; b3.x = w0[col + 48]; b3.y = w1[col + 48];
    accL0 = __builtin_amdgcn_wmma_f32_16x16x4_f32(false, aL, false, b0, (short)0, accL0, false, false);
    accL1 = __builtin_amdgcn_wmma_f32_16x16x4_f32(false, aL, false, b1, (short)0, accL1, false, false);
    accL2 = __builtin_amdgcn_wmma_f32_16x16x4_f32(false, aL, false, b2, (short)0, accL2, false, false);
    accL3 = __builtin_amdgcn_wmma_f32_16x16x4_f32(false, aL, false, b3, (short)0, accL3, false, false);
    accH0 = __builtin_amdgcn_wmma_f32_16x16x4_f32(false, aH, false, b0, (short)0, accH0, false, false);
    accH1 = __builtin_amdgcn_wmma_f32_16x16x4_f32(false, aH, false, b1, (short)0, accH1, false, false);
    accH2 = __builtin_amdgcn_wmma_f32_16x16x4_f32(false, aH, false, b2, (short)0, accH2, false, false);
    accH3 = __builtin_amdgcn_wmma_f32_16x16x4_f32(false, aH, false, b3, (short)0, accH3, false, false);
  }
  // C/D layout: VGPR r -> lanes 0-15: M=r, lanes 16-31: M=8+r; N = lane&15
  const int mHi = (lane >> 4) << 3;
  float* dLo = D + (size_t)(nodeBase + mHi) * Nc + colBase + row;
  float* dHi = dLo + (size_t)16 * Nc;
  #pragma unroll
  for (int r = 0; r < 8; ++r) {
    float* drL = dLo + (size_t)r * Nc;
    float* drH = dHi + (size_t)r * Nc;
    drL[0]  = accL0[r];
    drL[16] = accL1[r];
    drL[32] = accL2[r];
    drL[48] = accL3[r];
    drH[0]  = accH0[r];
    drH[16] = accH1[r];
    drH[32] = accH2[r];
    drH[48] = accH3[r];
  }
}

// ---------------------------------------------------------------- gather-propagate + bias + ReLU
// transposed=0: out[node*C + c] ; transposed=1: out[(b*C + c)*NN + n] (BCHW final)
__global__ void propagate_relu(const float* __restrict__ h, const int* __restrict__ rowptr,
                               const int* __restrict__ eSrc, const float* __restrict__ eW,
                               const float* __restrict__ bias, float* __restrict__ out,
                               int C, int transposed) {
  int idx = blockIdx.x * blockDim.x + threadIdx.x;
  int cq = C >> 2;
  int node = idx / cq;
  int c4 = (idx - node * cq) * 4;
  if (node >= NTOT) return;
  float4 acc = make_float4(0.f, 0.f, 0.f, 0.f);
  int e0 = rowptr[node], e1 = rowptr[node + 1];
  for (int e = e0; e < e1; ++e) {
    int s = eSrc[e];
    float w = eW[e];
    const float4 hv = *(const float4*)(h + (size_t)s * C + c4);
    acc.x += w * hv.x; acc.y += w * hv.y; acc.z += w * hv.z; acc.w += w * hv.w;
  }
  acc.x = fmaxf(acc.x + bias[c4 + 0], 0.f);
  acc.y = fmaxf(acc.y + bias[c4 + 1], 0.f);
  acc.z = fmaxf(acc.z + bias[c4 + 2], 0.f);
  acc.w = fmaxf(acc.w + bias[c4 + 3], 0.f);
  if (!transposed) {
    *(float4*)(out + (size_t)node * C + c4) = acc;
  } else {
    int b = node / NN, n = node - b * NN;
    size_t base = ((size_t)b * C) * NN + n;
    out[base + (size_t)(c4 + 0) * NN] = acc.x;
    out[base + (size_t)(c4 + 1) * NN] = acc.y;
    out[base + (size_t)(c4 + 2) * NN] = acc.z;
    out[base + (size_t)(c4 + 3) * NN] = acc.w;
  }
}

// ---------------------------------------------------------------- launch
extern "C" void kernel_launch(void* const* d_in, const int* in_sizes, int n_in,
                              void* d_out, int out_size, void* d_ws, size_t ws_size,
                              hipStream_t stream) {
  const float* dens = (const float*)d_in[0];   // [4,1,64,64]
  const float* fm   = (const float*)d_in[1];   // [4,256,64,64]  (channel-major = col-major A)
  const float* W1   = (const float*)d_in[2];   // [256,512]
  const float* b1   = (const float*)d_in[3];   // [512]
  const float* W2   = (const float*)d_in[4];   // [512,256]
  const float* b2   = (const float*)d_in[5];   // [256]
  float* out = (float*)d_out;                  // [4,256,64,64]

  char* cur = (char*)d_ws;
  auto carve = [&](size_t bytes) { void* p = (void*)cur; cur += (bytes + 255) & ~(size_t)255; return p; };
  int*   tgt    = (int*)  carve((size_t)NTOT * KNN * sizeof(int));
  int*   indeg  = (int*)  carve((size_t)NTOT * sizeof(int));
  int*   rowptr = (int*)  carve((size_t)(NTOT + 1) * sizeof(int));
  int*   cursor = (int*)  carve((size_t)NTOT * sizeof(int));
  float* dinv   = (float*)carve((size_t)NTOT * sizeof(float));
  int*   eSrc   = (int*)  carve((size_t)ETOT * sizeof(int));
  float* eW     = (float*)carve((size_t)ETOT * sizeof(float));
  float* h      = (float*)carve((size_t)NTOT * CHID * sizeof(float));
  float* hr     = (float*)carve((size_t)NTOT * CHID * sizeof(float));
  float* h2     = (float*)carve((size_t)NTOT * COUT * sizeof(float));

  // 1) graph construction
  init_deg_kernel<<<NTOT / 256, 256, 0, stream>>>(indeg, cursor);
  knn_kernel<<<NTOT / 256, 256, 0, stream>>>(dens, tgt, indeg);
  scan_kernel<<<1, 1024, 0, stream>>>(indeg, rowptr);
  dinv_kernel<<<NTOT / 256, 256, 0, stream>>>(indeg, dinv);
  fill_kernel<<<NTOT / 256, 256, 0, stream>>>(tgt, rowptr, cursor, dinv, eSrc, eW);

  // 2) layer 1: h = X @ W1 (fp32 WMMA, A col-major from BCHW), propagate + b1 + ReLU
  gemm32x64_wmma<true><<<NTOT / 32, 32 * (CHID / 64), 0, stream>>>(fm, W1, h, CIN, CHID);
  propagate_relu<<<(NTOT * (CHID / 4)) / 256, 256, 0, stream>>>(h, rowptr, eSrc, eW, b1, hr, CHID, 0);

  // 3) layer 2: h2 = hr @ W2 (A row-major), propagate + b2 + ReLU, write transposed BCHW
  gemm32x64_wmma<false><<<NTOT / 32, 32 * (COUT / 64), 0, stream>>>(hr, W2, h2, CHID, COUT);
  propagate_relu<<<(NTOT * (COUT / 4)) / 256, 256, 0, stream>>>(h2, rowptr, eSrc, eW, b2, out, COUT, 1);
}